// PointNetSetAbstraction_11484742550033
// MI455X (gfx1250) — compile-verified
//
#include <hip/hip_runtime.h>
#include <hip/hip_bf16.h>

// ---------------------------------------------------------------------------
// PointNet++ SetAbstraction for gfx1250 (MI455X), wave32.
//  K1: farthest point sampling (LDS-resident, shuffle argmax)
//  K2: ball query (first-32-in-radius scan) + new_xyz output
//  K3: gather + 3x [GEMM(V_WMMA_F32_16X16X4_F32) + LayerNorm + ReLU] + maxpool
// ---------------------------------------------------------------------------

#define B_    8
#define N_    4096
#define S_    1024
#define NS_   32
#define R2_   0.04f
#define EPS_  1e-5f
#define CPB_  4     // centers per block in MLP kernel

typedef float v2f __attribute__((ext_vector_type(2)));
typedef float v8f __attribute__((ext_vector_type(8)));

// ------------------------------ K1: FPS ------------------------------------
__global__ __launch_bounds__(1024) void fps_kernel(const float* __restrict__ xyz,
                                                   int* __restrict__ fps_idx) {
  const int b = blockIdx.x;
  const int t = threadIdx.x;
  __shared__ float px[N_], py[N_], pz[N_];
  __shared__ float dist[N_];
  __shared__ float wval[32];
  __shared__ int   widx[32];
  __shared__ int   s_far;

  const float* base = xyz + (size_t)b * N_ * 3;
#pragma unroll
  for (int i = 0; i < 4; i++) {
    int n = t + i * 1024;
    px[n] = base[3 * n + 0];
    py[n] = base[3 * n + 1];
    pz[n] = base[3 * n + 2];
    dist[n] = 1e10f;
  }
  __syncthreads();

  const int lane = t & 31, wid = t >> 5;
  int far = 0;
  for (int s = 0; s < S_; s++) {
    if (t == 0) fps_idx[b * S_ + s] = far;           // scan emits incoming carry
    float cx = px[far], cy = py[far], cz = pz[far];
    float bv = -1.0f;
    int   bi = 0x7fffffff;
#pragma unroll
    for (int i = 0; i < 4; i++) {
      int n = t + i * 1024;
      float dx = px[n] - cx, dy = py[n] - cy, dz = pz[n] - cz;
      float d = dx * dx + dy * dy + dz * dz;
      float dn = fminf(dist[n], d);
      dist[n] = dn;
      if (dn > bv || (dn == bv && n < bi)) { bv = dn; bi = n; }
    }
    // wave32 argmax reduction (first-index tiebreak)
#pragma unroll
    for (int off = 16; off > 0; off >>= 1) {
      float ov = __shfl_down(bv, off, 32);
      int   oi = __shfl_down(bi, off, 32);
      if (ov > bv || (ov == bv && oi < bi)) { bv = ov; bi = oi; }
    }
    if (lane == 0) { wval[wid] = bv; widx[wid] = bi; }
    __syncthreads();
    if (wid == 0) {
      bv = wval[lane]; bi = widx[lane];
#pragma unroll
      for (int off = 16; off > 0; off >>= 1) {
        float ov = __shfl_down(bv, off, 32);
        int   oi = __shfl_down(bi, off, 32);
        if (ov > bv || (ov == bv && oi < bi)) { bv = ov; bi = oi; }
      }
      if (lane == 0) s_far = bi;
    }
    __syncthreads();
    far = s_far;
  }
}

// --------------------------- K2: ball query --------------------------------
__global__ __launch_bounds__(256) void ballquery_kernel(const float* __restrict__ xyz,
                                                        const int* __restrict__ fps_idx,
                                                        int* __restrict__ gidx,
                                                        float* __restrict__ out_newxyz) {
  const int b = blockIdx.y;
  const int s = blockIdx.x * 256 + threadIdx.x;
  __shared__ float px[N_], py[N_], pz[N_];
  const float* base = xyz + (size_t)b * N_ * 3;
  for (int i = threadIdx.x; i < N_; i += 256) {
    px[i] = base[3 * i + 0];
    py[i] = base[3 * i + 1];
    pz[i] = base[3 * i + 2];
  }
  __syncthreads();

  const int f = fps_idx[b * S_ + s];
  const float cx = px[f], cy = py[f], cz = pz[f];
  out_newxyz[((size_t)b * S_ + s) * 3 + 0] = cx;
  out_newxyz[((size_t)b * S_ + s) * 3 + 1] = cy;
  out_newxyz[((size_t)b * S_ + s) * 3 + 2] = cz;

  int* g = gidx + ((size_t)b * S_ + s) * NS_;
  int cnt = 0;
  int first = f;  // center itself is always in radius, but track explicitly
  for (int n = 0; n < N_ && cnt < NS_; n++) {
    float dx = px[n] - cx, dy = py[n] - cy, dz = pz[n] - cz;
    float d = dx * dx + dy * dy + dz * dz;
    if (!(d > R2_)) {               // keep if d <= r^2 (matches reference mask)
      if (cnt == 0) first = n;
      g[cnt++] = n;
    }
  }
  for (int k = cnt; k < NS_; k++) g[k] = first;
}

// --------------------------- K3: grouped MLP -------------------------------
// sX stride 68 (bank-friendly: 68 mod 64 = 4), sH stride 129 (odd).
__device__ __forceinline__ void gemm_tile(const float* __restrict__ sX,
                                          const float* __restrict__ sW, int Nw, int K,
                                          float* __restrict__ sH,
                                          const float* __restrict__ bias,
                                          int m0, int n0, int lane) {
  const int mrow  = m0 + (lane & 15);
  const int khalf = (lane >> 4) << 1;     // 0 or 2
  const int ncol  = n0 + (lane & 15);
  v8f acc = {};
  for (int k = 0; k < K; k += 4) {
    v2f a, bb;
    a.x  = sX[mrow * 68 + k + khalf];
    a.y  = sX[mrow * 68 + k + khalf + 1];
    bb.x = sW[(k + khalf) * Nw + ncol];
    bb.y = sW[(k + khalf + 1) * Nw + ncol];
    acc = __builtin_amdgcn_wmma_f32_16x16x4_f32(false, a, false, bb,
                                                (short)0, acc, false, false);
  }
  const float bn = bias[ncol];
#pragma unroll
  for (int r = 0; r < 8; r++) {
    int row = m0 + r + ((lane >> 4) << 3);
    sH[row * 129 + ncol] = acc[r] + bn;
  }
}

__device__ __forceinline__ void layernorm_relu(float* __restrict__ sH, int C,
                                               const float* __restrict__ g,
                                               const float* __restrict__ be,
                                               float* __restrict__ dst, int dstride,
                                               float* __restrict__ sMu,
                                               float* __restrict__ sRs, int tid) {
  if (tid < 32) {
    float sum = 0.f;
    for (int c = 0; c < C; c++) sum += sH[tid * 129 + c];
    float mu = sum / (float)C;
    float sq = 0.f;
    for (int c = 0; c < C; c++) { float d = sH[tid * 129 + c] - mu; sq += d * d; }
    sMu[tid] = mu;
    sRs[tid] = rsqrtf(sq / (float)C + EPS_);
  }
  __syncthreads();
  for (int e = tid; e < 32 * C; e += 256) {
    int m = e / C, c = e - m * C;           // C is 64 or 128 -> cheap
    float v = sH[m * 129 + c];
    v = (v - sMu[m]) * sRs[m] * g[c] + be[c];
    dst[m * dstride + c] = fmaxf(v, 0.f);
  }
  __syncthreads();
}

__global__ __launch_bounds__(256) void mlp_kernel(
    const float* __restrict__ xyz, const float* __restrict__ points,
    const float* __restrict__ W1, const float* __restrict__ b1,
    const float* __restrict__ g1, const float* __restrict__ be1,
    const float* __restrict__ W2, const float* __restrict__ b2,
    const float* __restrict__ g2, const float* __restrict__ be2,
    const float* __restrict__ W3, const float* __restrict__ b3,
    const float* __restrict__ g3, const float* __restrict__ be3,
    const int* __restrict__ gidx, const float* __restrict__ newxyz,
    float* __restrict__ out_points) {
  __shared__ float sW1[68 * 64];      // [k][n], rows 67 zero-padded
  __shared__ float sW2[64 * 64];
  __shared__ float sW3[64 * 128];
  __shared__ float sPar[768];         // b1 g1 be1 | b2 g2 be2 | b3 g3 be3
  __shared__ float sX[32 * 68];
  __shared__ float sH[32 * 129];
  __shared__ float sMu[32], sRs[32];
  __shared__ int   sIdx[32];
  __shared__ float sCtr[3];

  const int tid = threadIdx.x;
  for (int i = tid; i < 67 * 64; i += 256) sW1[i] = W1[i];
  for (int i = tid; i < 64;      i += 256) sW1[67 * 64 + i] = 0.f;
  for (int i = tid; i < 64 * 64; i += 256) sW2[i] = W2[i];
  for (int i = tid; i < 64 * 128; i += 256) sW3[i] = W3[i];
  if (tid < 64) {
    sPar[tid]       = b1[tid];  sPar[64 + tid]  = g1[tid];  sPar[128 + tid] = be1[tid];
    sPar[192 + tid] = b2[tid];  sPar[256 + tid] = g2[tid];  sPar[320 + tid] = be2[tid];
  }
  if (tid < 128) {
    sPar[384 + tid] = b3[tid];  sPar[512 + tid] = g3[tid];  sPar[640 + tid] = be3[tid];
  }
  __syncthreads();

  const int lane = tid & 31, wid = tid >> 5;
  const int mt = (wid >> 2) * 16;       // 0 or 16
  const int nt = (wid & 3) * 16;        // 0,16,32,48

  for (int cc = 0; cc < CPB_; cc++) {
    const int cid = blockIdx.x * CPB_ + cc;
    const int b = cid >> 10;            // S_ = 1024
    if (tid < 32) sIdx[tid] = gidx[(size_t)cid * NS_ + tid];
    if (tid < 3)  sCtr[tid] = newxyz[(size_t)cid * 3 + tid];
    __syncthreads();

    // gather: combined[32][68] = [xyz - center (3) | points (64) | pad (1)]
    for (int e = tid; e < 32 * 68; e += 256) {
      int m = e / 68, c = e - m * 68;
      int n = sIdx[m];
      float v;
      if (c < 3)       v = xyz[((size_t)b * N_ + n) * 3 + c] - sCtr[c];
      else if (c < 67) v = points[((size_t)b * N_ + n) * 64 + (c - 3)];
      else             v = 0.f;
      sX[e] = v;
    }
    __syncthreads();

    // layer 1: (32x68) @ (68x64), 8 tiles -> 1 tile/wave
    gemm_tile(sX, sW1, 64, 68, sH, sPar + 0, mt, nt, lane);
    __syncthreads();
    layernorm_relu(sH, 64, sPar + 64, sPar + 128, sX, 68, sMu, sRs, tid);

    // layer 2: (32x64) @ (64x64)
    gemm_tile(sX, sW2, 64, 64, sH, sPar + 192, mt, nt, lane);
    __syncthreads();
    layernorm_relu(sH, 64, sPar + 256, sPar + 320, sX, 68, sMu, sRs, tid);

    // layer 3: (32x64) @ (64x128), 16 tiles -> 2 tiles/wave
    gemm_tile(sX, sW3, 128, 64, sH, sPar + 384, mt, nt, lane);
    gemm_tile(sX, sW3, 128, 64, sH, sPar + 384, mt, nt + 64, lane);
    __syncthreads();
    layernorm_relu(sH, 128, sPar + 512, sPar + 640, sH, 129, sMu, sRs, tid);

    // max over the 32 samples
    if (tid < 128) {
      float mx = -1e30f;
#pragma unroll 8
      for (int m = 0; m < 32; m++) mx = fmaxf(mx, sH[m * 129 + tid]);
      out_points[(size_t)cid * 128 + tid] = mx;
    }
    __syncthreads();
  }
}

// ------------------------------- launch ------------------------------------
extern "C" void kernel_launch(void* const* d_in, const int* in_sizes, int n_in,
                              void* d_out, int out_size, void* d_ws, size_t ws_size,
                              hipStream_t stream) {
  const float* xyz    = (const float*)d_in[0];
  const float* points = (const float*)d_in[1];
  const float* W1  = (const float*)d_in[2];
  const float* b1  = (const float*)d_in[3];
  const float* g1  = (const float*)d_in[4];
  const float* be1 = (const float*)d_in[5];
  const float* W2  = (const float*)d_in[6];
  const float* b2  = (const float*)d_in[7];
  const float* g2  = (const float*)d_in[8];
  const float* be2 = (const float*)d_in[9];
  const float* W3  = (const float*)d_in[10];
  const float* b3  = (const float*)d_in[11];
  const float* g3  = (const float*)d_in[12];
  const float* be3 = (const float*)d_in[13];

  float* out        = (float*)d_out;
  float* out_newxyz = out;                      // (8,1024,3)
  float* out_points = out + (size_t)B_ * S_ * 3; // (8,1024,128)

  int* fps  = (int*)d_ws;                               // 8*1024 ints
  int* gidx = (int*)((char*)d_ws + (size_t)B_ * S_ * 4); // 8*1024*32 ints

  fps_kernel<<<B_, 1024, 0, stream>>>(xyz, fps);
  ballquery_kernel<<<dim3(S_ / 256, B_), 256, 0, stream>>>(xyz, fps, gidx, out_newxyz);
  mlp_kernel<<<(B_ * S_) / CPB_, 256, 0, stream>>>(
      xyz, points, W1, b1, g1, be1, W2, b2, g2, be2, W3, b3, g3, be3,
      gidx, out_newxyz, out_points);
}